// GroupAttention_one_dim_68659347193906
// MI455X (gfx1250) — compile-verified
//
#include <hip/hip_runtime.h>
#include <math.h>

// Problem constants: B=2, S=256, D=256
#define BQ 2
#define SQ 256
#define DQ 256
#define EPS 1e-6f

typedef __attribute__((ext_vector_type(2))) float v2f;
typedef __attribute__((ext_vector_type(8))) float v8f;

// ---------------------------------------------------------------------------
// Kernel 0: prep.
//  Blocks 0..7 : pack the WMMA B-fragments for all 64 K-chunks into bpack.
//      B (4x16 f32) per chunk kc: col0 = g*w1, col1 = g*w2, col2 = ones,
//      cols 3..15 = 0. Lane L supplies rows K = 2*(L>>4), 2*(L>>4)+1 of its
//      column n = L&15. bpack[(kc*32 + lane)*2 + {0,1}].
//  Block 8     : LDS tree-reduce the 4 layernorm-fold constants:
//      consts = { SUM g*w1, SUM g*w2, SUM b*w1, SUM b*w2 }.
// ---------------------------------------------------------------------------
__global__ void __launch_bounds__(256)
prep(const float* __restrict__ ln_g, const float* __restrict__ ln_b,
     const float* __restrict__ fc_w, float* __restrict__ bpack,
     float* __restrict__ consts) {
  if (blockIdx.x < 8) {
    const int tid  = blockIdx.x * 256 + threadIdx.x;  // 0..2047 = (kc,lane)
    const int lane = tid & 31;
    const int kc   = tid >> 5;
    const int n    = lane & 15;
    const int bi   = kc * 4 + 2 * (lane >> 4);
    v2f pk;
    if (n == 0)      { pk.x = ln_g[bi] * fc_w[bi];
                       pk.y = ln_g[bi + 1] * fc_w[bi + 1]; }
    else if (n == 1) { pk.x = ln_g[bi] * fc_w[DQ + bi];
                       pk.y = ln_g[bi + 1] * fc_w[DQ + bi + 1]; }
    else if (n == 2) { pk.x = 1.0f; pk.y = 1.0f; }   // ones column -> row sum
    else             { pk.x = 0.0f; pk.y = 0.0f; }
    *(v2f*)(bpack + (size_t)tid * 2) = pk;
  } else {
    const int t = threadIdx.x;
    __shared__ float sh[4][256];
    const float g = ln_g[t], bb = ln_b[t];
    const float w1 = fc_w[t], w2 = fc_w[DQ + t];
    sh[0][t] = g * w1;  sh[1][t] = g * w2;
    sh[2][t] = bb * w1; sh[3][t] = bb * w2;
    __syncthreads();
    for (int off = 128; off >= 1; off >>= 1) {
      if (t < off) {
        sh[0][t] += sh[0][t + off];
        sh[1][t] += sh[1][t + off];
        sh[2][t] += sh[2][t + off];
        sh[3][t] += sh[3][t + off];
      }
      __syncthreads();
    }
    if (t == 0) {
      consts[0] = sh[0][0]; consts[1] = sh[1][0];
      consts[2] = sh[2][0]; consts[3] = sh[3][0];
    }
  }
}

// ---------------------------------------------------------------------------
// Kernel 1: per-row layernorm-folded dots via V_WMMA_F32_16X16X4_F32.
// One wave handles 16 rows; 64 chained WMMAs sweep K=256. Inner loop is
// branch-free: A b64 load + packed-B b64 load + 2 FMA (sum-of-squares) + WMMA.
// A (16x4 f32): lane m<16 supplies (M=m, K=0,1); lane m+16 supplies (K=2,3).
// D (16x16 f32): row m in acc[m%8], lanes (m/8)*16 + n.
// ---------------------------------------------------------------------------
__global__ void __launch_bounds__(32)
ln_dots_wmma(const float* __restrict__ ctx,
             const float* __restrict__ bpack,
             const float* __restrict__ consts,
             float* __restrict__ a_out,
             float* __restrict__ q_out) {
  const int lane = threadIdx.x;      // 0..31
  const int m    = lane & 15;
  const int half = lane >> 4;
  const int r0   = blockIdx.x * 16;  // 16 rows per wave

  const float ps1 = consts[0], ps2 = consts[1];
  const float pb1 = consts[2], pb2 = consts[3];

  const float* Arow = ctx + (size_t)(r0 + m) * DQ + 2 * half;
  const float* Brow = bpack + (size_t)lane * 2;

  v8f acc = {};
  float ssq = 0.f;
#pragma unroll 8
  for (int kc = 0; kc < DQ / 4; ++kc) {
    v2f afrag = *(const v2f*)(Arow + kc * 4);    // c[row][4kc+2h .. +1]
    v2f bfrag = *(const v2f*)(Brow + kc * 64);   // packed weight fragment
    ssq = fmaf(afrag.x, afrag.x, ssq);
    ssq = fmaf(afrag.y, afrag.y, ssq);
    acc = __builtin_amdgcn_wmma_f32_16x16x4_f32(
        /*neg_a=*/false, afrag, /*neg_b=*/false, bfrag,
        /*c_mod=*/(short)0, acc, /*reuse_a=*/false, /*reuse_b=*/false);
  }

  // sum of squares for row m = lane partial + partner-lane partial
  const float ssq_tot = ssq + __shfl_xor(ssq, 16, 32);

  // ---- extract D-matrix columns 0 (x.w1), 1 (x.w2), 2 (row sum) ----
  float rd1 = 0.f, rd2 = 0.f, rds = 0.f;
#define XTR(mm)                                                             \
  {                                                                         \
    float t1 = __shfl(acc[(mm) & 7], (((mm) >> 3) << 4) + 0, 32);           \
    float t2 = __shfl(acc[(mm) & 7], (((mm) >> 3) << 4) + 1, 32);           \
    float t3 = __shfl(acc[(mm) & 7], (((mm) >> 3) << 4) + 2, 32);           \
    if (lane == (mm)) { rd1 = t1; rd2 = t2; rds = t3; }                     \
  }
  XTR(0)  XTR(1)  XTR(2)  XTR(3)  XTR(4)  XTR(5)  XTR(6)  XTR(7)
  XTR(8)  XTR(9)  XTR(10) XTR(11) XTR(12) XTR(13) XTR(14) XTR(15)
#undef XTR

  if (lane < 16) {
    const int row  = r0 + lane;
    const float mu = rds * (1.0f / (float)DQ);
    float var = (ssq_tot - (float)DQ * mu * mu) * (1.0f / (float)(DQ - 1));
    var = var < 0.f ? 0.f : var;
    const float inv = 1.0f / (sqrtf(var) + EPS);
    a_out[row] = (rd1 - mu * ps1) * inv + pb1;   // dot(x_row, w1)
    q_out[row] = (rd2 - mu * ps2) * inv + pb2;   // dot(x_row, w2)
  }
}

// ---------------------------------------------------------------------------
// Kernel 2: per-batch inclusive sum scan of q -> exclusive prefix p[b][0..S]
// ---------------------------------------------------------------------------
__global__ void __launch_bounds__(SQ)
scan_q(const float* __restrict__ q, float* __restrict__ p) {
  const int b = blockIdx.x, t = threadIdx.x;
  __shared__ float sh[SQ];
  float v = q[b * SQ + t];
  sh[t] = v;
  __syncthreads();
  for (int off = 1; off < SQ; off <<= 1) {
    float o = (t >= off) ? sh[t - off] : 0.f;
    __syncthreads();
    v += o;
    sh[t] = v;
    __syncthreads();
  }
  p[b * (SQ + 1) + t + 1] = v;
  if (t == 0) p[b * (SQ + 1)] = 0.f;
}

// ---------------------------------------------------------------------------
// Kernel 3: per (b,i) block: Mij row, LDS product scan for G, symmetric store.
//   logit(k>=i) = a[k] + (p[k+1]-p[i])/(k-i+1) + fc_b ; else fc_b
//   Mij = prior + (1-prior)*sigmoid(logit)
//   G[i][k>=i] = prod_{j=i..k} Mij[i][j]; G symmetric; diag forced to 1.
// ---------------------------------------------------------------------------
__global__ void __launch_bounds__(SQ)
build_G(const float* __restrict__ a, const float* __restrict__ p,
        const float* __restrict__ prior, const float* __restrict__ fcb_p,
        float* __restrict__ out) {
  const int i = blockIdx.x, b = blockIdx.y, k = threadIdx.x;
  const float pr  = prior[0];
  const float fcb = fcb_p[0];
  const bool valid = (k >= i);

  const float a_k = a[b * SQ + k];
  const float pk1 = p[b * (SQ + 1) + k + 1];
  const float pi  = p[b * (SQ + 1) + i];
  const float logit = valid ? (a_k + (pk1 - pi) / (float)(k - i + 1) + fcb)
                            : fcb;  // cm==0 -> c1==0 -> only bias survives
  const float M = pr + (1.0f - pr) / (1.0f + expf(-logit));

  // second output: Mij, concatenated after G
  float* Mout = out + (size_t)BQ * SQ * SQ;
  Mout[((size_t)b * SQ + i) * SQ + k] = M;

  // inclusive product scan of Gfill = valid ? M : 1
  __shared__ float sh[SQ];
  float v = valid ? M : 1.0f;
  sh[k] = v;
  __syncthreads();
  for (int off = 1; off < SQ; off <<= 1) {
    float o = (k >= off) ? sh[k - off] : 1.f;
    __syncthreads();
    v *= o;
    sh[k] = v;
    __syncthreads();
  }

  float* G = out;
  const size_t base = (size_t)b * SQ * SQ;
  if (k > i) {
    G[base + (size_t)i * SQ + k] = v;   // upper triangle
    G[base + (size_t)k * SQ + i] = v;   // mirrored lower triangle
  } else if (k == i) {
    G[base + (size_t)i * SQ + i] = 1.0f;  // diag = 1 after reference's fixup
  }
}

// ---------------------------------------------------------------------------
extern "C" void kernel_launch(void* const* d_in, const int* in_sizes, int n_in,
                              void* d_out, int out_size, void* d_ws, size_t ws_size,
                              hipStream_t stream) {
  const float* ctx   = (const float*)d_in[0];  // (B,S,D)
  // d_in[1] = eos_mask (unused by reference)
  const float* prior = (const float*)d_in[2];  // (1,)
  const float* ln_g  = (const float*)d_in[3];  // (D,)
  const float* ln_b  = (const float*)d_in[4];  // (D,)
  const float* fc_w  = (const float*)d_in[5];  // (2D,)
  const float* fc_b  = (const float*)d_in[6];  // (1,)
  float* out = (float*)d_out;                  // [G | Mij], each B*S*S

  float* ws     = (float*)d_ws;
  float* bpack  = ws;                          // 64*32*2 = 4096 floats (16 KB)
  float* a_ws   = ws + 4096;                   // B*S = 512
  float* q_ws   = ws + 4096 + BQ * SQ;         // B*S = 512
  float* p_ws   = ws + 4096 + 2 * BQ * SQ;     // B*(S+1) = 514
  float* consts = ws + 4096 + 2 * BQ * SQ + BQ * (SQ + 1);  // 4

  prep<<<9, 256, 0, stream>>>(ln_g, ln_b, fc_w, bpack, consts);
  ln_dots_wmma<<<(BQ * SQ) / 16, 32, 0, stream>>>(ctx, bpack, consts,
                                                  a_ws, q_ws);
  scan_q<<<BQ, SQ, 0, stream>>>(q_ws, p_ws);
  build_G<<<dim3(SQ, BQ), SQ, 0, stream>>>(a_ws, p_ws, prior, fc_b, out);
}